// GCN_53094385713941
// MI455X (gfx1250) — compile-verified
//
#include <hip/hip_runtime.h>
#include <hip/hip_bf16.h>
#include <math.h>

// -------------------------------------------------------------------------
// GCN forward (2x GCNConv + ELU) for MI455X / gfx1250 (wave32, WMMA).
//  N=50000, D=128, E=800000. Aggregation is L2-resident (25.6MB per feature
//  matrix << 192MB L2); GEMMs use V_WMMA_F32_16X16X4_F32 (full fp32).
// -------------------------------------------------------------------------

typedef float v2f __attribute__((ext_vector_type(2)));
typedef float v8f __attribute__((ext_vector_type(8)));

#define DFEAT 128

// deg[i] = 1.0 (self-loop contribution)
__global__ void gcn_init_deg(float* __restrict__ deg, int N) {
    int i = blockIdx.x * blockDim.x + threadIdx.x;
    if (i < N) deg[i] = 1.0f;
}

// deg[dst[e]] += 1
__global__ void gcn_deg(const int* __restrict__ dst, float* __restrict__ deg, int E) {
    int e = blockIdx.x * blockDim.x + threadIdx.x;
    if (e < E) atomicAdd(&deg[dst[e]], 1.0f);
}

// deg -> rsqrt(deg) in place (deg >= 1 always, self-loops included)
__global__ void gcn_dinv(float* __restrict__ deg, int N) {
    int i = blockIdx.x * blockDim.x + threadIdx.x;
    if (i < N) deg[i] = rsqrtf(deg[i]);
}

// O[M x 128] = X[M x 128] @ W[128 x 128], fp32 WMMA 16x16x4.
// 256 threads = 8 waves; block handles 16 rows, wave w handles cols 16w..16w+15.
__global__ __launch_bounds__(256) void gcn_gemm_wmma(
    const float* __restrict__ X, const float* __restrict__ W,
    float* __restrict__ O, int M)
{
    __shared__ float As[16][132];   // stride 132: 132 % 64 = 4 -> bank-conflict-free

    const int tid  = threadIdx.x;
    const int lane = tid & 31;
    const int wave = tid >> 5;
    const int rowbase = blockIdx.x * 16;

    // Cooperative load of 16x128 fp32 A-tile (8KB) as float4, fully coalesced.
    {
        const float4* Xv = (const float4*)(X + (size_t)rowbase * DFEAT);
        #pragma unroll
        for (int idx = tid; idx < 512; idx += 256) {
            int r  = idx >> 5;        // 32 float4 per row
            int c4 = (idx & 31) << 2;
            float4 v = make_float4(0.f, 0.f, 0.f, 0.f);
            if (rowbase + r < M) v = Xv[idx];
            *(float4*)&As[r][c4] = v;
        }
    }
    __syncthreads();

    // Fragment layout (ISA 7.12.2, 32-bit A 16x4):
    //   lanes 0-15: M=lane, VGPR0=K0, VGPR1=K1 ; lanes 16-31: M=lane-16, K2/K3.
    const int row  = lane & 15;
    const int koff = (lane >> 4) << 1;
    const int cb   = wave << 4;          // column tile base
    const int cl   = lane & 15;          // column within tile (B/C/D lane = N)

    v8f acc = {};
    #pragma unroll
    for (int t = 0; t < 32; ++t) {       // K = 128 in steps of 4
        int k0 = (t << 2) + koff;
        v2f a = *(const v2f*)&As[row][k0];               // A[row][k0], A[row][k0+1]
        v2f b;
        b.x = W[(size_t)k0 * DFEAT + cb + cl];           // B[k0][col]
        b.y = W[(size_t)(k0 + 1) * DFEAT + cb + cl];     // B[k0+1][col]
        acc = __builtin_amdgcn_wmma_f32_16x16x4_f32(
            /*neg_a=*/false, a, /*neg_b=*/false, b,
            /*c_mod=*/(short)0, acc, /*reuse_a=*/false, /*reuse_b=*/false);
    }

    // C/D layout: lanes 0-15 -> rows 0..7 in VGPR 0..7; lanes 16-31 -> rows 8..15.
    // Per VGPR store: lanes 0-15 write 64 contiguous bytes (coalesced pair).
    const int row0 = rowbase + ((lane >> 4) << 3);
    const int col  = cb + cl;
    #pragma unroll
    for (int r = 0; r < 8; ++r) {
        int gr = row0 + r;
        if (gr < M) O[(size_t)gr * DFEAT + col] = acc[r];
    }
}

// acc[i][:] = xw[i][:] * dinv[i]^2   (self-loop term; also clears poison)
__global__ void gcn_self_init(const float* __restrict__ xw, const float* __restrict__ dinv,
                              float* __restrict__ acc, int n4) {
    int idx = blockIdx.x * blockDim.x + threadIdx.x;   // over N*32 float4s
    if (idx >= n4) return;
    int i = idx >> 5;
    float s = dinv[i]; s = s * s;
    float4 v = ((const float4*)xw)[idx];
    v.x *= s; v.y *= s; v.z *= s; v.w *= s;
    ((float4*)acc)[idx] = v;
}

// One wave per edge: acc[dst] += xw[src] * (dinv[src]*dinv[dst]).
// Lane loads a contiguous float4 -> 512B coalesced gather; 4 f32 L2 atomics.
__global__ __launch_bounds__(256) void gcn_scatter(
    const float* __restrict__ xw, const int* __restrict__ src,
    const int* __restrict__ dst, const float* __restrict__ dinv,
    float* __restrict__ acc, int E)
{
    int e = blockIdx.x * 8 + (threadIdx.x >> 5);
    if (e >= E) return;
    int lane = threadIdx.x & 31;
    int s  = src[e];
    int d0 = dst[e];
    float norm = dinv[s] * dinv[d0];
    float4 v = ((const float4*)(xw + (size_t)s * DFEAT))[lane];
    float* out = acc + (size_t)d0 * DFEAT + (lane << 2);
    atomicAdd(out + 0, v.x * norm);
    atomicAdd(out + 1, v.y * norm);
    atomicAdd(out + 2, v.z * norm);
    atomicAdd(out + 3, v.w * norm);
}

__device__ __forceinline__ float elu1(float x) {
    return x > 0.0f ? x : expm1f(x);
}

// acc = elu(acc + bias) in place
__global__ void gcn_elu_bias(float* __restrict__ acc, const float* __restrict__ bias, int n4) {
    int idx = blockIdx.x * blockDim.x + threadIdx.x;   // over N*32 float4s
    if (idx >= n4) return;
    int d4 = (idx & 31) << 2;
    float4 b = *(const float4*)&bias[d4];
    float4 v = ((const float4*)acc)[idx];
    v.x = elu1(v.x + b.x);
    v.y = elu1(v.y + b.y);
    v.z = elu1(v.z + b.z);
    v.w = elu1(v.w + b.w);
    ((float4*)acc)[idx] = v;
}

extern "C" void kernel_launch(void* const* d_in, const int* in_sizes, int n_in,
                              void* d_out, int out_size, void* d_ws, size_t ws_size,
                              hipStream_t stream) {
    const float* x   = (const float*)d_in[0];
    const int*   ei  = (const int*)  d_in[1];   // [2, E] row-major: src then dst
    const float* W1  = (const float*)d_in[2];
    const float* b1  = (const float*)d_in[3];
    const float* W2  = (const float*)d_in[4];
    const float* b2  = (const float*)d_in[5];
    float*       out = (float*)d_out;

    const int N = in_sizes[0] / DFEAT;
    const int E = in_sizes[1] / 2;
    const int* src = ei;
    const int* dst = ei + E;

    // Workspace layout: deg/dinv (N f32) | bufA (N*128 f32) | bufB (N*128 f32)
    char* ws = (char*)d_ws;
    size_t off = 0;
    float* deg  = (float*)(ws + off); off += ((size_t)N * 4 + 255) & ~(size_t)255;
    float* bufA = (float*)(ws + off); off += ((size_t)N * DFEAT * 4 + 255) & ~(size_t)255;
    float* bufB = (float*)(ws + off);

    const int n4 = N * (DFEAT / 4);             // float4 count per feature matrix
    dim3 blk(256);
    dim3 grN((N + 255) / 256);
    dim3 grE((E + 255) / 256);
    dim3 gr4((n4 + 255) / 256);
    dim3 grG((N + 15) / 16);                    // GEMM: one block per 16 rows
    dim3 grS((E + 7) / 8);                      // scatter: 8 edges per 256-thr block

    // Normalization: dinv = rsqrt(in-degree incl. self-loop)
    gcn_init_deg<<<grN, blk, 0, stream>>>(deg, N);
    gcn_deg     <<<grE, blk, 0, stream>>>(dst, deg, E);
    gcn_dinv    <<<grN, blk, 0, stream>>>(deg, N);

    // Layer 1: xw1 = x @ W1 ; aggregate ; h1 = elu(. + b1)
    gcn_gemm_wmma<<<grG, blk, 0, stream>>>(x, W1, bufA, N);
    gcn_self_init<<<gr4, blk, 0, stream>>>(bufA, deg, bufB, n4);
    gcn_scatter  <<<grS, blk, 0, stream>>>(bufA, src, dst, deg, bufB, E);
    gcn_elu_bias <<<gr4, blk, 0, stream>>>(bufB, b1, n4);

    // Layer 2: xw2 = h1 @ W2 ; aggregate into d_out ; out = elu(. + b2)
    gcn_gemm_wmma<<<grG, blk, 0, stream>>>(bufB, W2, bufA, N);
    gcn_self_init<<<gr4, blk, 0, stream>>>(bufA, deg, out, n4);
    gcn_scatter  <<<grS, blk, 0, stream>>>(bufA, src, dst, deg, out, E);
    gcn_elu_bias <<<gr4, blk, 0, stream>>>(out, b2, n4);
}